// MultiHeadAttention_9122510536719
// MI455X (gfx1250) — compile-verified
//
#include <hip/hip_runtime.h>
#include <hip/hip_bf16.h>

// ---------------------------------------------------------------------------
// MHA forward for MI455X (gfx1250), fp32 end-to-end via V_WMMA_F32_16X16X4_F32.
// out = [B,S,E] output ++ [B,H,S,S] attn (reference returns a tuple).
//
// Pipeline (all WMMA f32, wave32, one 16x16 tile per wave):
//   0) wtrans        : transpose Wq/Wk/Wv/Wo -> Wt[n][k] (float2 B loads)
//   1) qkv_proj      : Q/K = X@W + b head-split [B,H,S,D]; V -> [B,H,D,S]
//   2) scores        : raw attn = (Q K^T)/8, Q tile register-resident
//   3) softmax_ctx   : fused row softmax (L2-resident 16-row slab)
//                      + normalized attn write + ctx = attn@V via WMMA
//   4) out_proj      : out = merge_heads(ctx)@Wo + bo
// ---------------------------------------------------------------------------

#define BATCH  4
#define SEQ    2048
#define EMBED  512
#define HEADS  8
#define HDIM   64
#define BH     (BATCH * HEADS)          // 32
#define BS     (BATCH * SEQ)            // 8192
#define PROJ_ELEMS ((size_t)BATCH * HEADS * SEQ * HDIM)   // 4,194,304 floats
#define W_ELEMS    ((size_t)EMBED * EMBED)                // 262,144 floats

typedef __attribute__((ext_vector_type(2))) float v2f;
typedef __attribute__((ext_vector_type(8))) float v8f;

// D(16x16,f32) = A(16x4,f32) x B(4x16,f32) + C   — wave32, one matrix per wave.
// A layout: lane&15 = M, lane>>4 selects K pair {2*hi, 2*hi+1} in {v0,v1}.
// B layout: lane&15 = N, same K-pair striping.
// C/D layout: VGPR r, lane l -> D[M = r + 8*(l>>4)][N = l&15].
__device__ __forceinline__ v8f wmma_f32(v2f a, v2f b, v8f c) {
  return __builtin_amdgcn_wmma_f32_16x16x4_f32(
      /*neg_a=*/false, a, /*neg_b=*/false, b,
      /*c_mod=*/(short)0, c, /*reuse_a=*/false, /*reuse_b=*/false);
}

// ---------------------------------------------------------------------------
// Kernel 0: 512x512 transpose, Wt[n*512 + k] = W[k*512 + n].
// grid = (16, 16, 4), block = (32, 8); LDS-tiled 32x32.
// ---------------------------------------------------------------------------
__global__ __launch_bounds__(256) void wtrans_kernel(
    const float* __restrict__ W0, const float* __restrict__ W1,
    const float* __restrict__ W2, const float* __restrict__ W3,
    float* __restrict__ T0, float* __restrict__ T1,
    float* __restrict__ T2, float* __restrict__ T3) {
  const int which = blockIdx.z;
  const float* W = (which == 0) ? W0 : (which == 1) ? W1
                 : (which == 2) ? W2 : W3;
  float* T = (which == 0) ? T0 : (which == 1) ? T1
           : (which == 2) ? T2 : T3;

  __shared__ float tile[32][33];
  const int tx = threadIdx.x, ty = threadIdx.y;
  const int x0 = blockIdx.x * 32;  // k base
  const int y0 = blockIdx.y * 32;  // n base

#pragma unroll
  for (int i = 0; i < 4; ++i) {
    const int k = x0 + tx;             // column within W row
    const int n = y0 + ty + i * 8;     // W row is k-index? read W[k][n]:
    tile[ty + i * 8][tx] = W[(size_t)(x0 + ty + i * 8) * EMBED + (y0 + tx)];
  }
  __syncthreads();
#pragma unroll
  for (int i = 0; i < 4; ++i) {
    // T[n][k]: n = y0+ty+i*8 ... write transposed
    T[(size_t)(y0 + ty + i * 8) * EMBED + (x0 + tx)] = tile[tx][ty + i * 8];
  }
}

// ---------------------------------------------------------------------------
// Kernel 1: fused QKV projection.  Y = X @ W + b.
// Q,K stored head-split [B,H,S,D]; V stored head-transposed [B,H,D,S].
// Weights pre-transposed: B load is contiguous float2.
// grid = (EMBED/16/4, BS/16, 3), block = (32,4); one 16x16 tile per wave.
// ---------------------------------------------------------------------------
__global__ __launch_bounds__(128) void qkv_proj_kernel(
    const float* __restrict__ q_in, const float* __restrict__ k_in,
    const float* __restrict__ v_in,
    const float* __restrict__ WqT, const float* __restrict__ WkT,
    const float* __restrict__ WvT,
    const float* __restrict__ bq, const float* __restrict__ bk,
    const float* __restrict__ bv,
    float* __restrict__ Qo, float* __restrict__ Ko, float* __restrict__ Vt) {
  const int lane = threadIdx.x;
  const int lo = lane & 15, hi = lane >> 4;
  const int tn = blockIdx.x * 4 + threadIdx.y;  // 0..31  (E tiles)
  const int tm = blockIdx.y;                    // 0..511 (BS tiles)
  const int which = blockIdx.z;

  const float* X    = (which == 0) ? q_in : (which == 1) ? k_in : v_in;
  const float* Wt   = (which == 0) ? WqT  : (which == 1) ? WkT  : WvT;
  const float* bias = (which == 0) ? bq   : (which == 1) ? bk   : bv;

  const int row = tm * 16 + lo;  // flattened b*S+s
  const int col = tn * 16 + lo;  // e in [0,512)
  const float* Arow = X + (size_t)row * EMBED + 2 * hi;
  const float* Brow = Wt + (size_t)col * EMBED + 2 * hi;  // Wt[n][k]

  v8f acc = {0, 0, 0, 0, 0, 0, 0, 0};
#pragma unroll 8
  for (int k = 0; k < EMBED; k += 4) {
    v2f a = *(const v2f*)(Arow + k);
    v2f bm = *(const v2f*)(Brow + k);
    acc = wmma_f32(a, bm, acc);
  }

  const float bb = bias[col];
  const int h = col >> 6, d = col & 63;
  if (which != 2) {
    float* Y = (which == 0) ? Qo : Ko;
#pragma unroll
    for (int r = 0; r < 8; ++r) {
      const int m = tm * 16 + r + 8 * hi;  // flattened b*S+s
      const int b = m >> 11, s = m & (SEQ - 1);
      Y[(((size_t)b * HEADS + h) * SEQ + s) * HDIM + d] = acc[r] + bb;
    }
  } else {  // V: store [B,H,D,S]
#pragma unroll
    for (int r = 0; r < 8; ++r) {
      const int m = tm * 16 + r + 8 * hi;
      const int b = m >> 11, s = m & (SEQ - 1);
      Vt[(((size_t)b * HEADS + h) * HDIM + d) * SEQ + s] = acc[r] + bb;
    }
  }
}

// ---------------------------------------------------------------------------
// Kernel 2: raw scores = (Q K^T) / sqrt(D), streamed to attn region.
// Q tile (16x64) held in 16 float2 VGPRs (A layout), reused over 8 key tiles.
// grid = (4, SEQ/16, BH), block = (32,4): 16 wave-slots x 8 key tiles = 128.
// ---------------------------------------------------------------------------
__global__ __launch_bounds__(128) void scores_kernel(
    const float* __restrict__ Q, const float* __restrict__ K,
    float* __restrict__ attn) {
  const int lane = threadIdx.x;
  const int lo = lane & 15, hi = lane >> 4;
  const int tq = blockIdx.y;                    // 0..127 query tile
  const int bh = blockIdx.z;                    // 0..31

  const float* Aq = Q + ((size_t)bh * SEQ + tq * 16 + lo) * HDIM + 2 * hi;
  v2f qreg[16];
#pragma unroll
  for (int j = 0; j < 16; ++j) qreg[j] = *(const v2f*)(Aq + 4 * j);

  const int ktbase = (blockIdx.x * 4 + threadIdx.y) * 8;  // 8 key tiles/wave
  float* outb = attn + (size_t)bh * SEQ * SEQ;

  for (int i = 0; i < 8; ++i) {
    const int tk = ktbase + i;
    const float* Bk = K + ((size_t)bh * SEQ + tk * 16 + lo) * HDIM + 2 * hi;

    v8f acc = {0, 0, 0, 0, 0, 0, 0, 0};
#pragma unroll
    for (int j = 0; j < 16; ++j) {
      v2f b = *(const v2f*)(Bk + 4 * j);
      acc = wmma_f32(qreg[j], b, acc);
    }

#pragma unroll
    for (int r = 0; r < 8; ++r) {
      const int m = tq * 16 + r + 8 * hi;
      const int n = tk * 16 + lo;
      outb[(size_t)m * SEQ + n] = acc[r] * 0.125f;  // 1/sqrt(64)
    }
  }
}

// ---------------------------------------------------------------------------
// Kernel 3: fused softmax + context.  Each wave owns a 16-row slab of attn
// (128 KB, L2-resident):
//   pass A: row max; pass B: exp in place + row sum;
//   pass C: WMMA K-loop: load exp, scale by 1/sum, write final attn,
//           same registers feed the A operand of ctx = attn @ V.
// V is [B,H,D,S] so the B operand is a contiguous float2.
// grid = (SEQ/16/4, BH), block = (32,4).
// ---------------------------------------------------------------------------
__global__ __launch_bounds__(128) void softmax_context_kernel(
    float* __restrict__ attn, const float* __restrict__ Vt,
    float* __restrict__ ctx) {
  const int lane = threadIdx.x;
  const int lo = lane & 15, hi = lane >> 4;
  const int tq = blockIdx.x * 4 + threadIdx.y;  // 0..127
  const int bh = blockIdx.y;                    // 0..31

  float* Arow = attn + ((size_t)bh * SEQ + tq * 16 + lo) * SEQ;  // row `lo`
  float* half = Arow + hi * (SEQ / 2);  // this lane's half of the row

  // pass A: row max
  float m = -3.402823466e38f;
  for (int j = 0; j < SEQ / 2; j += 4) {
    float4 x = *(const float4*)(half + j);
    m = fmaxf(m, fmaxf(fmaxf(x.x, x.y), fmaxf(x.z, x.w)));
  }
  m = fmaxf(m, __shfl_xor(m, 16, 32));  // combine the two half-rows

  // pass B: exp in place + row sum
  float s = 0.0f;
  for (int j = 0; j < SEQ / 2; j += 4) {
    float4 x = *(const float4*)(half + j);
    x.x = __expf(x.x - m);
    x.y = __expf(x.y - m);
    x.z = __expf(x.z - m);
    x.w = __expf(x.w - m);
    s += (x.x + x.y) + (x.z + x.w);
    *(float4*)(half + j) = x;
  }
  s += __shfl_xor(s, 16, 32);
  const float inv = 1.0f / s;  // lane holds 1/rowsum for row `lo`

  // pass C: normalize + attn write + ctx WMMA accumulate
  const float* Vb0 = Vt + ((size_t)bh * HDIM + lo +  0) * SEQ + 2 * hi;
  const float* Vb1 = Vt + ((size_t)bh * HDIM + lo + 16) * SEQ + 2 * hi;
  const float* Vb2 = Vt + ((size_t)bh * HDIM + lo + 32) * SEQ + 2 * hi;
  const float* Vb3 = Vt + ((size_t)bh * HDIM + lo + 48) * SEQ + 2 * hi;

  v8f acc0 = {0, 0, 0, 0, 0, 0, 0, 0};
  v8f acc1 = {0, 0, 0, 0, 0, 0, 0, 0};
  v8f acc2 = {0, 0, 0, 0, 0, 0, 0, 0};
  v8f acc3 = {0, 0, 0, 0, 0, 0, 0, 0};

#pragma unroll 4
  for (int k = 0; k < SEQ; k += 4) {
    const int k0 = k + 2 * hi;
    v2f a = *(const v2f*)(Arow + k0);
    a.x *= inv;
    a.y *= inv;
    *(v2f*)(Arow + k0) = a;  // final normalized attn element (written once)

    v2f b0 = *(const v2f*)(Vb0 + k);
    v2f b1 = *(const v2f*)(Vb1 + k);
    v2f b2 = *(const v2f*)(Vb2 + k);
    v2f b3 = *(const v2f*)(Vb3 + k);
    acc0 = wmma_f32(a, b0, acc0);
    acc1 = wmma_f32(a, b1, acc1);
    acc2 = wmma_f32(a, b2, acc2);
    acc3 = wmma_f32(a, b3, acc3);
  }

  float* Cb = ctx + (size_t)bh * SEQ * HDIM;
#pragma unroll
  for (int r = 0; r < 8; ++r) {
    const int mm = tq * 16 + r + 8 * hi;
    float* crow = Cb + (size_t)mm * HDIM + lo;
    crow[0]  = acc0[r];
    crow[16] = acc1[r];
    crow[32] = acc2[r];
    crow[48] = acc3[r];
  }
}

// ---------------------------------------------------------------------------
// Kernel 4: output = merge_heads(ctx) @ Wo + bo   -> [B,S,E].
// Wo pre-transposed: contiguous float2 B loads.
// grid = (EMBED/16/4, BS/16), block = (32,4).
// ---------------------------------------------------------------------------
__global__ __launch_bounds__(128) void out_proj_kernel(
    const float* __restrict__ ctx, const float* __restrict__ WoT,
    const float* __restrict__ bo, float* __restrict__ out) {
  const int lane = threadIdx.x;
  const int lo = lane & 15, hi = lane >> 4;
  const int tn = blockIdx.x * 4 + threadIdx.y;  // 0..31
  const int tm = blockIdx.y;                    // 0..511

  const int row = tm * 16 + lo;  // flattened b*S+s
  const int col = tn * 16 + lo;  // e
  const int b = row >> 11, s = row & (SEQ - 1);
  // ctx[b, h, s, d]; A[m][k] gathers k = h*64+d
  const float* Cbase = ctx + (((size_t)b * HEADS) * SEQ + s) * HDIM;
  const float* Brow = WoT + (size_t)col * EMBED + 2 * hi;

  v8f acc = {0, 0, 0, 0, 0, 0, 0, 0};
#pragma unroll 8
  for (int k = 0; k < EMBED; k += 4) {
    const int k0 = k + 2 * hi;          // even, so {k0,k0+1} share head h
    const int h = k0 >> 6, d = k0 & 63;
    v2f a = *(const v2f*)(Cbase + (size_t)h * SEQ * HDIM + d);
    v2f bm = *(const v2f*)(Brow + k);
    acc = wmma_f32(a, bm, acc);
  }

  const float bb = bo[col];
#pragma unroll
  for (int r = 0; r < 8; ++r) {
    const int m = tm * 16 + r + 8 * hi;
    out[(size_t)m * EMBED + col] = acc[r] + bb;
  }
}

// ---------------------------------------------------------------------------
extern "C" void kernel_launch(void* const* d_in, const int* in_sizes, int n_in,
                              void* d_out, int out_size, void* d_ws,
                              size_t ws_size, hipStream_t stream) {
  const float* query = (const float*)d_in[0];
  const float* key   = (const float*)d_in[1];
  const float* value = (const float*)d_in[2];
  const float* Wq    = (const float*)d_in[3];
  const float* bq    = (const float*)d_in[4];
  const float* Wk    = (const float*)d_in[5];
  const float* bk    = (const float*)d_in[6];
  const float* Wv    = (const float*)d_in[7];
  const float* bv    = (const float*)d_in[8];
  const float* Wo    = (const float*)d_in[9];
  const float* bo    = (const float*)d_in[10];

  float* out  = (float*)d_out;                          // [B,S,E]
  float* attn = out + (size_t)BATCH * SEQ * EMBED;      // [B,H,S,S]

  float* Qw  = (float*)d_ws;           // [B,H,S,D]
  float* Kw  = Qw + PROJ_ELEMS;        // [B,H,S,D]
  float* Vtw = Kw + PROJ_ELEMS;        // [B,H,D,S]
  float* Cw  = Vtw + PROJ_ELEMS;       // context [B,H,S,D]
  float* WqT = Cw + PROJ_ELEMS;        // transposed weights
  float* WkT = WqT + W_ELEMS;
  float* WvT = WkT + W_ELEMS;
  float* WoT = WvT + W_ELEMS;

  dim3 blk(32, 4);

  // 0) weight transposes
  wtrans_kernel<<<dim3(16, 16, 4), dim3(32, 8), 0, stream>>>(
      Wq, Wk, Wv, Wo, WqT, WkT, WvT, WoT);

  // 1) QKV projections (z picks tensor; V stored [B,H,D,S])
  qkv_proj_kernel<<<dim3(EMBED / 16 / 4, BS / 16, 3), blk, 0, stream>>>(
      query, key, value, WqT, WkT, WvT, bq, bk, bv, Qw, Kw, Vtw);

  // 2) raw scaled scores -> attn region (Q register-resident)
  scores_kernel<<<dim3(4, SEQ / 16, BH), blk, 0, stream>>>(Qw, Kw, attn);

  // 3) fused softmax + normalized attn write + context = attn @ V
  softmax_context_kernel<<<dim3(SEQ / 16 / 4, BH), blk, 0, stream>>>(
      attn, Vtw, Cw);

  // 4) output projection
  out_proj_kernel<<<dim3(EMBED / 16 / 4, BS / 16), blk, 0, stream>>>(
      Cw, WoT, bo, out);
}